// CPCLoss_42657615183937
// MI455X (gfx1250) — compile-verified
//
#include <hip/hip_runtime.h>
#include <hip/hip_bf16.h>

typedef float v2f __attribute__((ext_vector_type(2)));
typedef float v8f __attribute__((ext_vector_type(8)));

#define SEQn 2048
#define Bn   64
#define Ln   128
#define ZDn  128
#define CDn  128
#define TLn  256   // 2*L

__global__ void zero_out_kernel(float* out) {
    if (threadIdx.x == 0) out[0] = 0.f;
}

__global__ __launch_bounds__(256) void cpc_loss_kernel(
    const float* __restrict__ z, const float* __restrict__ c,
    const int* __restrict__ mask, const int* __restrict__ negmap,
    const float* __restrict__ W, const float* __restrict__ lnw,
    const float* __restrict__ lnb, float* __restrict__ out)
{
    extern __shared__ float smem[];
    float* sZ = smem;                    // 256*128 floats (later aliased as sM)
    float* sC = smem + TLn * ZDn;        // 128*128 floats (c_t, row-major [j][c])
    float* sX = sC + Ln * CDn;           // 128*128 floats (X = W @ c_t^T)
    unsigned* sBits = (unsigned*)sX;     // 2048-bit mask bitmap (freed before sX written)

    const int b    = blockIdx.x;
    const int tid  = threadIdx.x;        // 256 threads = 8 waves (wave32)
    const int lane = tid & 31;
    const int wave = tid >> 5;
    const int l15  = lane & 15;
    const int klo  = (lane >> 4) << 1;   // 0 or 2 (K sub-offset per WMMA layout)

    const v8f vzero = {0.f,0.f,0.f,0.f,0.f,0.f,0.f,0.f};

    // ---- Phase 1: bitmap of masked rows for this batch column ----
    if (tid < 64) sBits[tid] = 0u;
    __syncthreads();
    if (tid < Ln) {
        int s = mask[tid * Bn + b];
        atomicOr(&sBits[s >> 5], 1u << (s & 31));
    }
    // gather c_t: two threads per row, float4 copies
    {
        int j = tid >> 1, h = tid & 1;
        int s = mask[j * Bn + b];
        const float4* src = (const float4*)(c + ((size_t)s * Bn + b) * CDn);
        float4* dst = (float4*)(sC + j * CDn);
        #pragma unroll
        for (int q = 0; q < 16; ++q) dst[h * 16 + q] = src[h * 16 + q];
    }
    __syncthreads();

    // ---- Phase 2: gather z_all (row tid), zeroing collided negatives ----
    {
        float4* dst = (float4*)(sZ + tid * ZDn);
        if (tid < Ln) {
            int s = mask[tid * Bn + b];
            const float4* src = (const float4*)(z + ((size_t)s * Bn + b) * ZDn);
            #pragma unroll 8
            for (int q = 0; q < 32; ++q) dst[q] = src[q];
        } else {
            int s = negmap[(tid - Ln) * Bn + b];
            bool iszero = (sBits[s >> 5] >> (s & 31)) & 1u;
            if (iszero) {
                float4 z4 = make_float4(0.f, 0.f, 0.f, 0.f);
                #pragma unroll 8
                for (int q = 0; q < 32; ++q) dst[q] = z4;
            } else {
                const float4* src = (const float4*)(z + ((size_t)s * Bn + b) * ZDn);
                #pragma unroll 8
                for (int q = 0; q < 32; ++q) dst[q] = src[q];
            }
        }
    }
    // ---- Phase 3: layernorm row tid in place (thread-private row, no barrier) ----
    {
        float* row = sZ + tid * ZDn;
        float mu = 0.f;
        for (int k = 0; k < ZDn; ++k) mu += row[k];
        mu *= (1.f / ZDn);
        float var = 0.f;
        for (int k = 0; k < ZDn; ++k) { float d = row[k] - mu; var += d * d; }
        var *= (1.f / ZDn);
        float rs = rsqrtf(var + 1e-5f);
        for (int k = 0; k < ZDn; ++k) row[k] = (row[k] - mu) * rs * lnw[k] + lnb[k];
    }
    __syncthreads();

    // ---- Phase 4 (GEMM-a, fp32 WMMA): X[zr][j] = sum_c W[zr][c] * c_t[j][c] ----
    // wave w owns row tile tm = w; W streamed from global (L2-resident, 64KB)
    {
        const int tm = wave;
        const int m  = tm * 16 + l15;
        for (int tn = 0; tn < 8; ++tn) {
            const int n = tn * 16 + l15;
            v8f acc = vzero;
            for (int k0 = 0; k0 < CDn; k0 += 4) {
                v2f a  = *(const v2f*)(W  + m * CDn + k0 + klo);
                v2f bb = *(const v2f*)(sC + n * CDn + k0 + klo);  // B[k][n] = c_t[n][k]
                acc = __builtin_amdgcn_wmma_f32_16x16x4_f32(
                          false, a, false, bb, (short)0, acc, false, false);
            }
            const int rbase = tm * 16 + ((lane >> 4) << 3);
            #pragma unroll
            for (int r = 0; r < 8; ++r) sX[(rbase + r) * Ln + n] = acc[r];
        }
    }
    __syncthreads();

    // ---- Phase 5 (GEMM-b, fp32 WMMA): M = sZ(256x128) @ sX(128x128), stored over sZ ----
    // wave w owns M rows [32w, 32w+32) == exactly the sZ rows it reads -> no cross-wave
    // hazard from aliasing; hold the 8-tile strip in registers until its K-loop finishes.
    float* sM = sZ;
    for (int tmi = 0; tmi < 2; ++tmi) {
        const int tm = wave * 2 + tmi;
        const int m  = tm * 16 + l15;
        v8f acc[8];
        #pragma unroll
        for (int t = 0; t < 8; ++t) acc[t] = vzero;
        for (int k0 = 0; k0 < ZDn; k0 += 4) {
            v2f a = *(const v2f*)(sZ + m * ZDn + k0 + klo);
            #pragma unroll
            for (int tn = 0; tn < 8; ++tn) {
                const int n = tn * 16 + l15;
                v2f bb;
                bb.x = sX[(k0 + klo    ) * Ln + n];
                bb.y = sX[(k0 + klo + 1) * Ln + n];
                acc[tn] = __builtin_amdgcn_wmma_f32_16x16x4_f32(
                              false, a, false, bb, (short)0, acc[tn], false, false);
            }
        }
        const int rbase = tm * 16 + ((lane >> 4) << 3);
        #pragma unroll
        for (int tn = 0; tn < 8; ++tn)
            #pragma unroll
            for (int r = 0; r < 8; ++r)
                sM[(rbase + r) * Ln + tn * 16 + l15] = acc[tn][r];
    }
    __syncthreads();

    // ---- Phase 6: column softmax stats + diagonal loss ----
    float contrib = 0.f;
    if (tid < Ln) {
        const int col = tid;
        float mx = -3.4e38f;
        for (int i = 0; i < TLn; ++i) mx = fmaxf(mx, sM[i * Ln + col]);
        float ssum = 0.f;
        for (int i = 0; i < TLn; ++i) ssum += expf(sM[i * Ln + col] - mx);
        float p = expf(sM[col * Ln + col] - mx) / ssum;
        contrib = -logf(p + 0.001f);
    }
    float* sRed = sC;
    sRed[tid] = contrib;
    __syncthreads();
    for (int off = 128; off > 0; off >>= 1) {
        if (tid < off) sRed[tid] += sRed[tid + off];
        __syncthreads();
    }
    if (tid == 0) atomicAdd(out, sRed[0] * (1.f / (Ln * Bn)));
}

extern "C" void kernel_launch(void* const* d_in, const int* in_sizes, int n_in,
                              void* d_out, int out_size, void* d_ws, size_t ws_size,
                              hipStream_t stream) {
    const float* z      = (const float*)d_in[0];
    const float* c      = (const float*)d_in[1];
    const int*   mask   = (const int*)  d_in[2];
    const int*   negmap = (const int*)  d_in[3];
    const float* W      = (const float*)d_in[4];
    const float* lnw    = (const float*)d_in[5];
    const float* lnb    = (const float*)d_in[6];
    float* out = (float*)d_out;

    zero_out_kernel<<<1, 32, 0, stream>>>(out);

    size_t smem = (size_t)(TLn * ZDn + Ln * CDn + Ln * Ln) * sizeof(float); // 256 KB
    cpc_loss_kernel<<<Bn, 256, smem, stream>>>(z, c, mask, negmap, W, lnw, lnb, out);
}